// HistogramMatcher_73254962201060
// MI455X (gfx1250) — compile-verified
//
#include <hip/hip_runtime.h>
#include <hip/hip_bf16.h>

// ---------------------------------------------------------------------------
// Histogram matching, 1024x1024x3 fp32 (HWC interleaved).
//   Phase 0: zero 6x256 u32 histograms in ws (ws is poisoned by harness).
//   Phase 1: LDS-privatized histograms (ds_add_u32) -> global merge.
//   Phase 2: one block, 6 waves: per-histogram 256-elem cumsum via
//            v_wmma_f32_16x16x4_f32 (exact in f32: counts <= 2^20 < 2^24),
//            normalize to CDF, then 768-entry pxmap interpolation.
//   Phase 3: per-pixel remap through uniform cs grid, float4 (B128) traffic.
// ---------------------------------------------------------------------------

#define NPIX   (1024 * 1024)
#define NELEM  (NPIX * 3)
#define NF4    (NELEM / 4)

typedef float v2f __attribute__((ext_vector_type(2)));
typedef float v8f __attribute__((ext_vector_type(8)));

// ------------------------------- Phase 0 -----------------------------------
__global__ void hm_zero_kernel(unsigned* __restrict__ gh) {
    int i = blockIdx.x * blockDim.x + threadIdx.x;
    if (i < 6 * 256) gh[i] = 0u;
}

// ------------------------------- Phase 1 -----------------------------------
__device__ __forceinline__ void hist_one(float v, unsigned* h) {
    float c = fminf(fmaxf(v, -1.0f), 1.0f);
    int bin = (int)((c + 1.0f) * 128.0f);   // trunc, matches astype(int32)
    bin = bin < 0 ? 0 : (bin > 255 ? 255 : bin);
    atomicAdd(&h[bin], 1u);                 // ds_add_u32
}

__global__ __launch_bounds__(256)
void hm_hist_kernel(const float* __restrict__ src, const float* __restrict__ tgt,
                    unsigned* __restrict__ gh) {
    __shared__ unsigned shh[6 * 256];
    for (int i = threadIdx.x; i < 6 * 256; i += blockDim.x) shh[i] = 0u;
    __syncthreads();

    const int stride = gridDim.x * blockDim.x;
    for (int i = blockIdx.x * blockDim.x + threadIdx.x; i < 2 * NF4; i += stride) {
        const int img = (i >= NF4) ? 1 : 0;
        const int ii  = img ? (i - NF4) : i;
        const float4 v = ((const float4*)(img ? tgt : src))[ii];  // B128 load
        const int c0 = ii % 3;                 // (4*ii)%3 == ii%3
        unsigned* base = &shh[img * 3 * 256];
        hist_one(v.x, base + c0 * 256);
        hist_one(v.y, base + ((c0 + 1) % 3) * 256);
        hist_one(v.z, base + ((c0 + 2) % 3) * 256);
        hist_one(v.w, base + c0 * 256);
    }
    __syncthreads();
    for (int i = threadIdx.x; i < 6 * 256; i += blockDim.x) {
        unsigned c = shh[i];
        if (c) atomicAdd(&gh[i], c);
    }
}

// ------------------------------- Phase 2 -----------------------------------
// One block, 192 threads = 6 waves; wave w owns histogram w (0..2 src, 3..5 tgt).
// cumsum(H) viewed as 16x16: P = H x U (U[k][c] = k<=c), then
// acc = P + L x W  (L[r][k] = k<r, W[k][*] = P[k][15]) -> full inclusive scan.
__global__ __launch_bounds__(192)
void hm_cdf_pxmap_kernel(const unsigned* __restrict__ gh, float* __restrict__ pxg) {
    __shared__ float histf[6][256];
    __shared__ float Pl[6][256];
    __shared__ float cdfl[6][256];
    __shared__ float csl[256];

    const int tid  = threadIdx.x;
    const int w    = tid >> 5;     // wave id == histogram id
    const int lane = tid & 31;
    const int N    = lane & 15;    // col for B/C/D frags, row M for A frags
    const int half = lane >> 4;

    // u32 counts -> f32 (exact, counts <= 2^20)
    #pragma unroll
    for (int q = 0; q < 8; ++q) {
        int idx = lane * 8 + q;
        histf[w][idx] = (float)gh[w * 256 + idx];
    }
    for (int j = tid; j < 256; j += 192)
        csl[j] = fminf(-1.0f + (float)j / 127.0f, 1.0f);
    __syncthreads();

    // P = H x U  (4 chained K=4 WMMAs)
    v8f acc = {0.f, 0.f, 0.f, 0.f, 0.f, 0.f, 0.f, 0.f};
    #pragma unroll
    for (int kb = 0; kb < 16; kb += 4) {
        const int k0 = kb + 2 * half;
        v2f a, b;
        a[0] = histf[w][N * 16 + k0];          // A[M][k], M = lane&15
        a[1] = histf[w][N * 16 + k0 + 1];
        b[0] = (k0     <= N) ? 1.0f : 0.0f;    // U[k][N]
        b[1] = (k0 + 1 <= N) ? 1.0f : 0.0f;
        acc = __builtin_amdgcn_wmma_f32_16x16x4_f32(false, a, false, b,
                                                    (short)0, acc, false, false);
    }
    #pragma unroll
    for (int v = 0; v < 8; ++v)
        Pl[w][(v + 8 * half) * 16 + N] = acc[v];   // D layout: row = v+8*half
    __syncthreads();

    // acc = P + L x W   (W[k][*] = P[k][15] = row-k running total)
    #pragma unroll
    for (int kb = 0; kb < 16; kb += 4) {
        const int k0 = kb + 2 * half;
        v2f a, b;
        a[0] = (k0     < N) ? 1.0f : 0.0f;     // L[M][k]
        a[1] = (k0 + 1 < N) ? 1.0f : 0.0f;
        b[0] = Pl[w][(k0    ) * 16 + 15];
        b[1] = Pl[w][(k0 + 1) * 16 + 15];
        acc = __builtin_amdgcn_wmma_f32_16x16x4_f32(false, a, false, b,
                                                    (short)0, acc, false, false);
    }

    // Normalize: (cdf - cdf[0]) * 2 / (npix - 1) - 1   (cdf_min == cdf[0])
    const float cdf0 = histf[w][0];
    #pragma unroll
    for (int v = 0; v < 8; ++v) {
        int idx = (v + 8 * half) * 16 + N;
        cdfl[w][idx] = (acc[v] - cdf0) * 2.0f / 1048575.0f - 1.0f;
    }
    __syncthreads();

    // pxmap[ch][i] = interpolate(dx = cdf_tgt[ch], dy = cs, x = cdf_src[ch][i])
    for (int e = tid; e < 768; e += 192) {
        const int ch = e >> 8, i = e & 255;
        const float x = cdfl[ch][i];
        const float* dx = cdfl[3 + ch];
        float best = fabsf(dx[0] - x);
        int i1 = 0;
        for (int j = 1; j < 256; ++j) {        // first-argmin, strict <
            float d = fabsf(dx[j] - x);
            if (d < best) { best = d; i1 = j; }
        }
        const int i0 = (i1 > 0) ? i1 - 1 : 0;
        const float x0 = dx[i0], x1 = dx[i1];
        const float den = x1 - x0;
        const float safe = (den == 0.0f) ? 1.0f : den;
        float r = csl[i0] + (csl[i1] - csl[i0]) * (x - x0) / safe;
        r = (x <= dx[0]) ? csl[0] : ((x >= dx[255]) ? csl[255] : r);
        pxg[ch * 256 + i] = r;
    }
}

// ------------------------------- Phase 3 -----------------------------------
// Nearest index on the uniform cs grid (cs[j] = min(-1 + j/127, 1);
// cs[254] == cs[255] == 1.0 so first-argmin is always <= 254).
__device__ __forceinline__ float map_px(float v, const float* __restrict__ cs,
                                        const float* __restrict__ pm) {
    const float t = (v + 1.0f) * 127.0f;
    int j0 = (int)floorf(t) - 1;
    j0 = j0 < 0 ? 0 : (j0 > 252 ? 252 : j0);
    int i1 = j0;
    float best = fabsf(cs[j0] - v);
    #pragma unroll
    for (int d = 1; d < 3; ++d) {              // window [j0, j0+2] <= 254
        float dd = fabsf(cs[j0 + d] - v);
        if (dd < best) { best = dd; i1 = j0 + d; }
    }
    const int i0 = (i1 > 0) ? i1 - 1 : 0;
    const float x0 = cs[i0], x1 = cs[i1];
    const float den = x1 - x0;
    const float safe = (den == 0.0f) ? 1.0f : den;
    const float r = pm[i0] + (pm[i1] - pm[i0]) * (v - x0) / safe;
    return (v <= -1.0f) ? pm[0] : ((v >= 1.0f) ? pm[255] : r);
}

__global__ __launch_bounds__(256)
void hm_map_kernel(const float* __restrict__ src, const float* __restrict__ pxg,
                   float* __restrict__ out) {
    __shared__ float pml[768];
    __shared__ float csl[256];
    for (int i = threadIdx.x; i < 768; i += blockDim.x) pml[i] = pxg[i];
    for (int j = threadIdx.x; j < 256; j += blockDim.x)
        csl[j] = fminf(-1.0f + (float)j / 127.0f, 1.0f);
    __syncthreads();

    const int stride = gridDim.x * blockDim.x;
    for (int i = blockIdx.x * blockDim.x + threadIdx.x; i < NF4; i += stride) {
        const float4 v = ((const float4*)src)[i];          // B128 load
        const int c0 = i % 3;
        float4 o;
        o.x = map_px(v.x, csl, pml + c0 * 256);
        o.y = map_px(v.y, csl, pml + ((c0 + 1) % 3) * 256);
        o.z = map_px(v.z, csl, pml + ((c0 + 2) % 3) * 256);
        o.w = map_px(v.w, csl, pml + c0 * 256);
        ((float4*)out)[i] = o;                             // B128 store
    }
}

// ------------------------------- Launch ------------------------------------
extern "C" void kernel_launch(void* const* d_in, const int* in_sizes, int n_in,
                              void* d_out, int out_size, void* d_ws, size_t ws_size,
                              hipStream_t stream) {
    const float* src = (const float*)d_in[0];
    const float* tgt = (const float*)d_in[1];
    float* out = (float*)d_out;

    unsigned* gh = (unsigned*)d_ws;                        // 6*256 u32  (6144 B)
    float* pxg = (float*)((char*)d_ws + 6 * 256 * sizeof(unsigned)); // 3*256 f32

    hm_zero_kernel<<<6, 256, 0, stream>>>(gh);
    hm_hist_kernel<<<1024, 256, 0, stream>>>(src, tgt, gh);
    hm_cdf_pxmap_kernel<<<1, 192, 0, stream>>>(gh, pxg);
    hm_map_kernel<<<1024, 256, 0, stream>>>(src, pxg, out);
}